// GCN_ori_55937654063396
// MI455X (gfx1250) — compile-verified
//
#include <hip/hip_runtime.h>
#include <math.h>

typedef float v2f __attribute__((ext_vector_type(2)));
typedef float v8f __attribute__((ext_vector_type(8)));

#define NUM_G 64
#define HDIM  128
#define CDIM  10

// ---------------------------------------------------------------------------
// degree / norm
// ---------------------------------------------------------------------------
__global__ __launch_bounds__(256) void init_deg_kernel(float* __restrict__ deg, long n) {
  long i = (long)blockIdx.x * blockDim.x + threadIdx.x;
  if (i < n) deg[i] = 1.0f;  // self-loop contributes 1 to every node's degree
}

__global__ __launch_bounds__(256) void zero_kernel(float* __restrict__ p, long n) {
  long stride = (long)gridDim.x * blockDim.x;
  for (long i = (long)blockIdx.x * blockDim.x + threadIdx.x; i < n; i += stride)
    p[i] = 0.0f;
}

__global__ __launch_bounds__(256) void deg_accum_kernel(const int* __restrict__ dst,
                                                        float* __restrict__ deg, long E) {
  long e = (long)blockIdx.x * blockDim.x + threadIdx.x;
  if (e < E) atomicAdd(&deg[dst[e]], 1.0f);
}

__global__ __launch_bounds__(256) void dinv_kernel(float* __restrict__ d, long n) {
  long i = (long)blockIdx.x * blockDim.x + threadIdx.x;
  if (i < n) {
    float v = d[i];
    d[i] = (v > 0.0f) ? rsqrtf(v) : 0.0f;
  }
}

// ---------------------------------------------------------------------------
// WMMA fp32 GEMM:  C[N,128] = A[N,128] @ W[128,128]
// grid.x = row tiles of 128, grid.y = 8 column tiles of 16.
// Block = 256 threads = 8 waves; each wave owns one 16x16 output tile.
// W column tile staged in LDS; A loaded as v2f (global_load_b64).
// ---------------------------------------------------------------------------
__global__ __launch_bounds__(256)
void gemm_wmma_kernel(const float* __restrict__ A, const float* __restrict__ W,
                      float* __restrict__ C, long N) {
  __shared__ float wtile[HDIM * 16];  // wtile[k*16 + j] = W[k][col0 + j]
  const int t    = threadIdx.x;
  const int col0 = blockIdx.y * 16;

  for (int i = t; i < HDIM * 16; i += 256) {
    int k = i >> 4, j = i & 15;
    wtile[i] = W[(long)k * HDIM + col0 + j];
  }
  __syncthreads();

  const int wave = t >> 5;
  const int lane = t & 31;
  const int half = lane >> 4;  // 0: lanes 0-15, 1: lanes 16-31
  const int lm   = lane & 15;

  const long row0 = (long)blockIdx.x * 128 + (long)wave * 16;
  long arow = row0 + lm;
  if (arow >= N) arow = N - 1;            // clamp address, keep EXEC all-ones
  const float* aptr = A + arow * HDIM + 2 * half;

  v8f acc = {};
#pragma unroll
  for (int k0 = 0; k0 < HDIM; k0 += 4) {
    // A: lane holds row (lane%16), K = k0 + 2*half + {0,1}  -> contiguous pair
    v2f a = *(const v2f*)(aptr + k0);
    // B: lane holds col (lane%16), K = k0 + 2*half + {0,1}
    v2f b;
    b.x = wtile[(k0 + 2 * half + 0) * 16 + lm];
    b.y = wtile[(k0 + 2 * half + 1) * 16 + lm];
    acc = __builtin_amdgcn_wmma_f32_16x16x4_f32(
        /*neg_a=*/false, a, /*neg_b=*/false, b,
        /*c_mod=*/(short)0, acc, /*reuse_a=*/false, /*reuse_b=*/false);
  }

  // D layout: VGPR v -> row = row0 + v + 8*half, col = col0 + lm
#pragma unroll
  for (int v = 0; v < 8; ++v) {
    long r = row0 + v + 8 * half;
    if (r < N) C[r * HDIM + col0 + lm] = acc[v];
  }
}

// ---------------------------------------------------------------------------
// Edge-parallel gather+scale+scatter-add.  One block per (edge or self-loop),
// 128 threads = features.  e >= E are the N self loops.
// ---------------------------------------------------------------------------
__global__ __launch_bounds__(128)
void scatter_kernel(const float* __restrict__ h, const float* __restrict__ dinv,
                    const int* __restrict__ ei, float* __restrict__ agg,
                    long E, long N) {
  long e = blockIdx.x;
  int f = threadIdx.x;
  int s, d;
  if (e < E) { s = ei[e]; d = ei[E + e]; }
  else       { s = d = (int)(e - E); }
  float w = dinv[s] * dinv[d];
  float v = h[(long)s * HDIM + f] * w;
  atomicAdd(&agg[(long)d * HDIM + f], v);
}

__global__ __launch_bounds__(256)
void bias_relu_kernel(float* __restrict__ agg, const float* __restrict__ b, long total) {
  long i = (long)blockIdx.x * blockDim.x + threadIdx.x;
  if (i < total) {
    float v = agg[i] + b[i & (HDIM - 1)];
    agg[i] = v > 0.0f ? v : 0.0f;
  }
}

// ---------------------------------------------------------------------------
// Pooling: per-block 64x128 LDS accumulator (ds_add_f32), then global flush.
// ---------------------------------------------------------------------------
__global__ __launch_bounds__(256)
void count_kernel(const int* __restrict__ batch, float* __restrict__ cnt, long N) {
  long i = (long)blockIdx.x * blockDim.x + threadIdx.x;
  if (i < N) atomicAdd(&cnt[batch[i]], 1.0f);
}

__global__ __launch_bounds__(256)
void pool_kernel(const float* __restrict__ h, const int* __restrict__ batch,
                 float* __restrict__ pooled, long N) {
  __shared__ float acc[NUM_G * HDIM];  // 32 KB of the 320 KB WGP LDS
  const int t = threadIdx.x;
  for (int i = t; i < NUM_G * HDIM; i += 256) acc[i] = 0.0f;
  __syncthreads();

  long chunk = (N + gridDim.x - 1) / gridDim.x;
  long start = (long)blockIdx.x * chunk;
  long end   = start + chunk; if (end > N) end = N;

  const int f     = t & (HDIM - 1);
  const int which = t >> 7;  // 0 or 1: two nodes processed per 256-thread pass
  for (long n = start + which; n < end; n += 2) {
    int g = batch[n];
    atomicAdd(&acc[g * HDIM + f], h[n * HDIM + f]);   // ds_add_f32
  }
  __syncthreads();

  for (int i = t; i < NUM_G * HDIM; i += 256)
    atomicAdd(&pooled[i], acc[i]);
}

// ---------------------------------------------------------------------------
// Final: mean (+b2 folded in), linear 128x10 (+bl), log_softmax per row.
// One block of 64 threads, one graph per thread.
// ---------------------------------------------------------------------------
__global__ __launch_bounds__(64)
void head_kernel(const float* __restrict__ pooled, const float* __restrict__ cnt,
                 const float* __restrict__ b2, const float* __restrict__ Wl,
                 const float* __restrict__ bl, float* __restrict__ out) {
  int g = threadIdx.x;
  if (g >= NUM_G) return;
  float inv = 1.0f / fmaxf(cnt[g], 1.0f);
  float logits[CDIM];
#pragma unroll
  for (int c = 0; c < CDIM; ++c) logits[c] = bl[c];
  for (int f = 0; f < HDIM; ++f) {
    float m = pooled[g * HDIM + f] * inv + b2[f];  // mean(v + b2) = mean(v) + b2
#pragma unroll
    for (int c = 0; c < CDIM; ++c) logits[c] += m * Wl[f * CDIM + c];
  }
  float mx = logits[0];
#pragma unroll
  for (int c = 1; c < CDIM; ++c) mx = fmaxf(mx, logits[c]);
  float s = 0.0f;
#pragma unroll
  for (int c = 0; c < CDIM; ++c) s += __expf(logits[c] - mx);
  float lse = __logf(s);
#pragma unroll
  for (int c = 0; c < CDIM; ++c) out[g * CDIM + c] = logits[c] - mx - lse;
}

// ---------------------------------------------------------------------------
// host-side launcher
// ---------------------------------------------------------------------------
extern "C" void kernel_launch(void* const* d_in, const int* in_sizes, int n_in,
                              void* d_out, int out_size, void* d_ws, size_t ws_size,
                              hipStream_t stream) {
  const float* x   = (const float*)d_in[0];
  const int*   ei  = (const int*)d_in[1];   // [2,E] flat: row0 = src, row1 = dst
  const int*   bat = (const int*)d_in[2];
  const float* W1  = (const float*)d_in[3];
  const float* b1  = (const float*)d_in[4];
  const float* W2  = (const float*)d_in[5];
  const float* b2  = (const float*)d_in[6];
  const float* Wl  = (const float*)d_in[7];
  const float* bl  = (const float*)d_in[8];
  float* out = (float*)d_out;

  const long N = (long)in_sizes[0] / HDIM;
  const long E = (long)in_sizes[1] / 2;
  const long Etot = E + N;                  // edges + self loops
  const long Npad = (N + 127) & ~127L;

  // workspace layout (floats)
  float* f      = (float*)d_ws;
  float* dinv   = f;                        // [Npad]
  float* h      = dinv + Npad;              // [Npad*128]
  float* agg    = h + Npad * HDIM;          // [Npad*128]
  float* pooled = agg + Npad * HDIM;        // [64*128]
  float* cnt    = pooled + NUM_G * HDIM;    // [64]

  const int T = 256;
  const long NH = N * HDIM;
  dim3 gemm_grid((unsigned)((N + 127) / 128), HDIM / 16);

  // --- normalization ---
  init_deg_kernel<<<(unsigned)((N + T - 1) / T), T, 0, stream>>>(dinv, N);
  deg_accum_kernel<<<(unsigned)((E + T - 1) / T), T, 0, stream>>>(ei + E, dinv, E);
  dinv_kernel<<<(unsigned)((N + T - 1) / T), T, 0, stream>>>(dinv, N);

  // --- conv1: h = x@W1 ; agg = scatter(norm * h[src]) ; relu(agg + b1) ---
  gemm_wmma_kernel<<<gemm_grid, T, 0, stream>>>(x, W1, h, N);
  zero_kernel<<<2048, T, 0, stream>>>(agg, NH);
  scatter_kernel<<<(unsigned)Etot, HDIM, 0, stream>>>(h, dinv, ei, agg, E, N);
  bias_relu_kernel<<<(unsigned)((NH + T - 1) / T), T, 0, stream>>>(agg, b1, NH);

  // --- conv2: h = agg@W2 ; agg = scatter(norm * h[src])  (b2 folded into head) ---
  gemm_wmma_kernel<<<gemm_grid, T, 0, stream>>>(agg, W2, h, N);
  zero_kernel<<<2048, T, 0, stream>>>(agg, NH);
  scatter_kernel<<<(unsigned)Etot, HDIM, 0, stream>>>(h, dinv, ei, agg, E, N);

  // --- mean pool + linear + log_softmax ---
  zero_kernel<<<32, T, 0, stream>>>(pooled, NUM_G * HDIM + NUM_G);
  count_kernel<<<(unsigned)((N + T - 1) / T), T, 0, stream>>>(bat, cnt, N);
  pool_kernel<<<64, T, 0, stream>>>(agg, bat, pooled, N);
  head_kernel<<<1, 64, 0, stream>>>(pooled, cnt, b2, Wl, bl, out);
}